// AltitudeConsistencyLoss_45672682225768
// MI455X (gfx1250) — compile-verified
//
#include <hip/hip_runtime.h>

typedef __attribute__((ext_vector_type(2))) float v2f;
typedef __attribute__((ext_vector_type(8))) float v8f;

#define L_LOC   8192
#define A_ALT   4
#define D_DIM   256
#define NSEG    (L_LOC * A_ALT)     /* 32768 segments = global row ids   */
#define B_ROWS  262144
#define NTILES  (NSEG / 16)         /* 2048 16-row Gram tiles            */
#define WPB     2                   /* waves per block in gram kernel    */
#define RSTRIDE 260                 /* LDS row stride (floats), 260%64=4 */

// ---------------------------------------------------------------------------
// workspace zero-init (sums + counts + 2 accumulators)
// ---------------------------------------------------------------------------
__global__ void k_zero(float* __restrict__ ws, int n) {
    int i = blockIdx.x * blockDim.x + threadIdx.x;
    int stride = gridDim.x * blockDim.x;
    for (; i < n; i += stride) ws[i] = 0.0f;
}

// ---------------------------------------------------------------------------
// phase 1: segment scatter-add. one wave32 per embedding row (1 KB),
// coalesced float4 reads, fire-and-forget f32 atomics resolved in L2.
// ---------------------------------------------------------------------------
__global__ __launch_bounds__(256) void k_scatter(const float* __restrict__ emb,
                                                 const int*   __restrict__ labels,
                                                 const int*   __restrict__ alts,
                                                 float*       __restrict__ sums,
                                                 unsigned*    __restrict__ cnts) {
    int gwave = (blockIdx.x * blockDim.x + threadIdx.x) >> 5;
    int lane  = threadIdx.x & 31;
    if (gwave >= B_ROWS) return;

    int row = gwave;
    int seg = labels[row] * A_ALT + (alts[row] / 50 - 3);  // 150,200,250,300 -> 0..3

    const float4* src = (const float4*)(emb + (size_t)row * D_DIM);
    float* dst = sums + (size_t)seg * D_DIM;

#pragma unroll
    for (int j = 0; j < 2; ++j) {
        int c4 = lane + j * 32;            // float4 index 0..63
        float4 v = src[c4];
        int c = c4 * 4;
        __hip_atomic_fetch_add(dst + c + 0, v.x, __ATOMIC_RELAXED, __HIP_MEMORY_SCOPE_AGENT);
        __hip_atomic_fetch_add(dst + c + 1, v.y, __ATOMIC_RELAXED, __HIP_MEMORY_SCOPE_AGENT);
        __hip_atomic_fetch_add(dst + c + 2, v.z, __ATOMIC_RELAXED, __HIP_MEMORY_SCOPE_AGENT);
        __hip_atomic_fetch_add(dst + c + 3, v.w, __ATOMIC_RELAXED, __HIP_MEMORY_SCOPE_AGENT);
    }
    if (lane == 0)
        __hip_atomic_fetch_add(&cnts[seg], 1u, __ATOMIC_RELAXED, __HIP_MEMORY_SCOPE_AGENT);
}

// ---------------------------------------------------------------------------
// phase 2: per 16-row tile (4 locations x 4 altitudes) compute the 16x16
// Gram matrix of the raw segment sums with V_WMMA_F32_16X16X4_F32 (K=256 in
// 64 steps, A and B fragments identical for M*M^T), then form cosines from
// Gram off-diagonals / sqrt(diag_i * diag_j) -- scale-invariance makes the
// mean-division and pre-normalization unnecessary.
// ---------------------------------------------------------------------------
__global__ __launch_bounds__(64) void k_gram(const float*    __restrict__ sums,
                                             const unsigned* __restrict__ cnts,
                                             float*          __restrict__ acc) {
    __shared__ float    lds[WPB][16 * RSTRIDE];
    __shared__ float    diag[WPB][16];
    __shared__ unsigned scc[WPB][16];

    int w    = threadIdx.x >> 5;
    int lane = threadIdx.x & 31;
    int tile = blockIdx.x * WPB + w;          // 0..NTILES-1
    float* T = lds[w];

    // stage 16 rows x 256 f32 (16 KB) coalesced into LDS, padded stride
    const float4* src = (const float4*)(sums + (size_t)tile * 16 * D_DIM);
#pragma unroll 4
    for (int it = 0; it < 32; ++it) {
        int idx = it * 32 + lane;             // float4 index within tile
        int r   = idx >> 6;
        int c4  = idx & 63;
        float4 v = src[idx];
        float* d = T + r * RSTRIDE + c4 * 4;
        d[0] = v.x; d[1] = v.y; d[2] = v.z; d[3] = v.w;
    }
    if (lane < 16) scc[w][lane] = cnts[tile * 16 + lane];
    __syncthreads();

    // 16x16x4 f32 A fragment: lane&15 = M, VGPR0 = K+ (lane<16?0:2), VGPR1 = +1
    int m    = lane & 15;
    int koff = (lane < 16) ? 0 : 2;
    const float* rowp = T + m * RSTRIDE + koff;

    v8f c = {};
#pragma unroll 8
    for (int kb = 0; kb < D_DIM; kb += 4) {
        v2f a;
        a.x = rowp[kb + 0];
        a.y = rowp[kb + 1];
        c = __builtin_amdgcn_wmma_f32_16x16x4_f32(
                /*neg_a=*/false, a, /*neg_b=*/false, a,
                /*c_mod=*/(short)0, c, /*reuse_a=*/false, /*reuse_b=*/false);
    }

    // extract diagonal: lanes 0-15 VGPR r -> G[r][lane]; lanes 16-31 -> G[r+8][lane-16]
#pragma unroll
    for (int r = 0; r < 8; ++r) {
        int M = r + ((lane < 16) ? 0 : 8);
        if (M == (lane & 15)) diag[w][M] = c[r];
    }
    __syncthreads();

    int   n  = lane & 15;
    bool  pn = scc[w][n] > 0u;
    float dn = diag[w][n];

    float totalLoc = 0.0f, cntLoc = 0.0f;
#pragma unroll
    for (int r = 0; r < 8; ++r) {
        int M = r + ((lane < 16) ? 0 : 8);
        bool sameLoc = (M >> 2) == (n >> 2);   // same location group of 4
        bool upper   = (M & 3) < (n & 3);      // altitude i < altitude j
        bool pm      = scc[w][M] > 0u;
        if (sameLoc && upper && pm && pn) {
            float denom = sqrtf(diag[w][M]) * sqrtf(dn);
            float cosv  = c[r] / fmaxf(denom, 1e-24f);
            totalLoc += 1.0f - cosv;
            cntLoc   += 1.0f;
        }
    }

    // wave32 reduction, then two atomics per wave
#pragma unroll
    for (int off = 16; off > 0; off >>= 1) {
        totalLoc += __shfl_down(totalLoc, off, 32);
        cntLoc   += __shfl_down(cntLoc,   off, 32);
    }
    if (lane == 0) {
        __hip_atomic_fetch_add(&acc[0], totalLoc, __ATOMIC_RELAXED, __HIP_MEMORY_SCOPE_AGENT);
        __hip_atomic_fetch_add(&acc[1], cntLoc,   __ATOMIC_RELAXED, __HIP_MEMORY_SCOPE_AGENT);
    }
}

// ---------------------------------------------------------------------------
// phase 3: finalize scalar loss
// ---------------------------------------------------------------------------
__global__ void k_final(const float* __restrict__ acc, float* __restrict__ out) {
    out[0] = (acc[1] > 0.0f) ? acc[0] / acc[1] : 0.0f;
}

// ---------------------------------------------------------------------------
extern "C" void kernel_launch(void* const* d_in, const int* in_sizes, int n_in,
                              void* d_out, int out_size, void* d_ws, size_t ws_size,
                              hipStream_t stream) {
    (void)in_sizes; (void)n_in; (void)out_size; (void)ws_size;

    const float* emb    = (const float*)d_in[0];
    const int*   labels = (const int*)d_in[1];
    const int*   alts   = (const int*)d_in[2];
    float*       out    = (float*)d_out;

    float*    sums = (float*)d_ws;
    unsigned* cnts = (unsigned*)(sums + (size_t)NSEG * D_DIM);
    float*    acc  = (float*)(cnts + NSEG);

    int nzero = NSEG * D_DIM + NSEG + 2;
    k_zero<<<2048, 256, 0, stream>>>((float*)d_ws, nzero);
    k_scatter<<<B_ROWS / 8, 256, 0, stream>>>(emb, labels, alts, sums, cnts);
    k_gram<<<NTILES / WPB, 32 * WPB, 0, stream>>>(sums, cnts, acc);
    k_final<<<1, 1, 0, stream>>>(acc, out);
}